// GNN_52664888983659
// MI455X (gfx1250) — compile-verified
//
#include <hip/hip_runtime.h>
#include <math.h>

typedef __attribute__((ext_vector_type(2))) float v2f;
typedef __attribute__((ext_vector_type(8))) float v8f;

#define F_IN   602
#define KTOT   (2 * F_IN)   // 1204
#define HDIM   256
#define K2TOT  (2 * HDIM)   // 512
#define NCLS   41
#define NB0    1024
#define NB1    25600

// ---------------------------------------------------------------------------
// Layer-0 fused GEMM: out[n x 256] = relu( concat(self, mean_NUM(neigh)) @ W )
// One workgroup = MT-row output tile x full N=256.
//   waves = MT/4 (wave w: row-group w>>2, col-group w&3, 16x64 tile each)
// A tile (MT x 64 K-chunk) staged in LDS with concat + neighbor-mean fused
// into the load, so the neighbor stream (x2) is read exactly once from HBM.
// Neighbor loads are non-temporal so the streaming 616MB doesn't evict W0
// from L2. fp32 WMMA 16x16x4 keeps full fp32 precision (bandwidth-bound).
// ---------------------------------------------------------------------------
template <int NUM, int MT>
__global__ __launch_bounds__(MT * 8) void sage_gemm_relu(
    const float* __restrict__ self,   // [n x 602]
    const float* __restrict__ neigh,  // [n*NUM x 602]
    const float* __restrict__ W,      // [1204 x 256]
    float* __restrict__ out)          // [n x 256]
{
    __shared__ __align__(16) float ldsA[MT][68];   // 64-wide K chunk + pad

    const int tid  = threadIdx.x;                  // MT*8 threads = MT/4 waves
    const int wave = tid >> 5;
    const int lane = tid & 31;
    const int m0   = blockIdx.x * MT;
    const float inv = 1.0f / (float)NUM;

    v8f acc0 = {}, acc1 = {}, acc2 = {}, acc3 = {};

    const int rowg = wave >> 2;                     // row-group (16 rows each)
    const int arow = (rowg << 4) + (lane & 15);     // local A row in LDS
    const int kh   = (lane >> 4) << 1;              // 0 or 2
    const int col0 = ((wave & 3) << 6) + (lane & 15);

    for (int kc = 0; kc < KTOT; kc += 64) {
        // ---- cooperative A-tile load: concat + neighbor mean fused ----
        __syncthreads();
#pragma unroll
        for (int p = 0; p < 8; ++p) {
            int e  = p * (MT * 8) + tid;
            int r  = e >> 6;
            int kk = e & 63;
            int k  = kc + kk;
            float v = 0.0f;
            if (k < KTOT) {
                long row = (long)(m0 + r);
                if (k < F_IN) {
                    v = self[row * F_IN + k];
                } else {
                    const float* nb = neigh + (row * NUM) * (long)F_IN + (k - F_IN);
                    float s = 0.0f;
#pragma unroll
                    for (int j = 0; j < NUM; ++j)
                        s += __builtin_nontemporal_load(&nb[(long)j * F_IN]);
                    v = s * inv;
                }
            }
            ldsA[r][kk] = v;
        }
        __syncthreads();

        const int krem   = KTOT - kc;
        const int ksteps = (krem < 64 ? krem : 64) >> 2;
        for (int s = 0; s < ksteps; ++s) {
            int ks = s << 2;
            v2f a = *(const v2f*)&ldsA[arow][ks + kh];
            int kr = kc + ks + kh;
            const float* Wp = W + (long)kr * HDIM + col0;
            v2f b0, b1, b2, b3;
            b0.x = Wp[0];  b0.y = Wp[HDIM + 0];
            b1.x = Wp[16]; b1.y = Wp[HDIM + 16];
            b2.x = Wp[32]; b2.y = Wp[HDIM + 32];
            b3.x = Wp[48]; b3.y = Wp[HDIM + 48];
            acc0 = __builtin_amdgcn_wmma_f32_16x16x4_f32(false, a, false, b0, (short)0, acc0, false, false);
            acc1 = __builtin_amdgcn_wmma_f32_16x16x4_f32(false, a, false, b1, (short)0, acc1, false, false);
            acc2 = __builtin_amdgcn_wmma_f32_16x16x4_f32(false, a, false, b2, (short)0, acc2, false, false);
            acc3 = __builtin_amdgcn_wmma_f32_16x16x4_f32(false, a, false, b3, (short)0, acc3, false, false);
        }
    }

    // ---- store with ReLU; C/D layout: row = i + 8*(lane>=16), col = lane%16 ----
    const int rbase = (rowg << 4) + ((lane >> 4) << 3);
#pragma unroll
    for (int i = 0; i < 8; ++i) {
        long row = (long)(m0 + rbase + i);
        float* o = out + row * HDIM + col0;
        o[0]  = fmaxf(acc0[i], 0.0f);
        o[16] = fmaxf(acc1[i], 0.0f);
        o[32] = fmaxf(acc2[i], 0.0f);
        o[48] = fmaxf(acc3[i], 0.0f);
    }
}

// ---------------------------------------------------------------------------
// agg2[1024 x 256] = mean over 25 consecutive rows of h1
// ---------------------------------------------------------------------------
__global__ __launch_bounds__(256) void mean25_kernel(
    const float* __restrict__ h1, float* __restrict__ agg2)
{
    int idx = blockIdx.x * 256 + threadIdx.x;     // r*256 + c
    int r = idx >> 8;
    int c = idx & 255;
    const float* p = h1 + ((long)r * 25) * HDIM + c;
    float s = 0.0f;
#pragma unroll
    for (int j = 0; j < 25; ++j) s += p[(long)j * HDIM];
    agg2[idx] = s * (1.0f / 25.0f);
}

// ---------------------------------------------------------------------------
// logits[1024 x 41] = concat(h0, agg2) @ W1   (WMMA, N padded to 48)
// Guarded columns are handled branchlessly (clamped address + select) so the
// inner WMMA loop has no EXEC manipulation.
// ---------------------------------------------------------------------------
__global__ __launch_bounds__(32) void dense_gemm(
    const float* __restrict__ h0, const float* __restrict__ agg2,
    const float* __restrict__ W,      // [512 x 41]
    float* __restrict__ logits)       // [1024 x 41]
{
    const int lane = threadIdx.x;
    const int m0   = blockIdx.x * 16;
    const int arow = lane & 15;
    const int kh   = (lane >> 4) << 1;
    const int col  = lane & 15;
    const int n2   = 32 + col;
    const bool ok2 = (n2 < NCLS);
    const int n2c  = ok2 ? n2 : (NCLS - 1);        // clamped, always in-bounds

    v8f acc0 = {}, acc1 = {}, acc2 = {};

    for (int k = 0; k < K2TOT; k += 4) {
        int k0 = k + kh;
        const float* Ab = (k0 < HDIM)
            ? (h0   + (long)(m0 + arow) * HDIM + k0)
            : (agg2 + (long)(m0 + arow) * HDIM + (k0 - HDIM));
        v2f a = *(const v2f*)Ab;

        const float* Wp = W + (long)k0 * NCLS;
        v2f b0, b1, b2;
        b0.x = Wp[col];       b0.y = Wp[NCLS + col];
        b1.x = Wp[16 + col];  b1.y = Wp[NCLS + 16 + col];
        float w2a = Wp[n2c], w2b = Wp[NCLS + n2c];
        b2.x = ok2 ? w2a : 0.0f;
        b2.y = ok2 ? w2b : 0.0f;

        acc0 = __builtin_amdgcn_wmma_f32_16x16x4_f32(false, a, false, b0, (short)0, acc0, false, false);
        acc1 = __builtin_amdgcn_wmma_f32_16x16x4_f32(false, a, false, b1, (short)0, acc1, false, false);
        acc2 = __builtin_amdgcn_wmma_f32_16x16x4_f32(false, a, false, b2, (short)0, acc2, false, false);
    }

    const int rbase = (lane >> 4) << 3;
#pragma unroll
    for (int i = 0; i < 8; ++i) {
        long row = (long)(m0 + rbase + i);
        float* o = logits + row * NCLS;
        o[col]      = acc0[i];
        o[16 + col] = acc1[i];
        if (ok2) o[n2] = acc2[i];
    }
}

// ---------------------------------------------------------------------------
// Row-wise log_softmax over 41 classes: one wave32 per row, shfl reductions.
// ---------------------------------------------------------------------------
__global__ __launch_bounds__(256) void logsoftmax_kernel(
    const float* __restrict__ logits, float* __restrict__ out)
{
    const int wave = threadIdx.x >> 5;
    const int lane = threadIdx.x & 31;
    const long row = (long)blockIdx.x * 8 + wave;

    const float* p = logits + row * NCLS;
    float v0 = p[lane];                         // lane 0..31 < 41 always valid
    bool  has2 = (lane + 32) < NCLS;
    float v1 = has2 ? p[lane + 32] : -INFINITY;

    float m = fmaxf(v0, v1);
    for (int off = 16; off > 0; off >>= 1)
        m = fmaxf(m, __shfl_xor(m, off, 32));

    float s = expf(v0 - m) + (has2 ? expf(v1 - m) : 0.0f);
    for (int off = 16; off > 0; off >>= 1)
        s += __shfl_xor(s, off, 32);

    float lse = m + logf(s);
    float* o = out + row * NCLS;
    o[lane] = v0 - lse;
    if (has2) o[lane + 32] = v1 - lse;
}

// ---------------------------------------------------------------------------
extern "C" void kernel_launch(void* const* d_in, const int* in_sizes, int n_in,
                              void* d_out, int out_size, void* d_ws, size_t ws_size,
                              hipStream_t stream)
{
    (void)in_sizes; (void)n_in; (void)out_size; (void)ws_size;

    const float* x0 = (const float*)d_in[0];   // [1024 x 602]
    const float* x1 = (const float*)d_in[1];   // [25600 x 602]
    const float* x2 = (const float*)d_in[2];   // [256000 x 602]
    const float* W0 = (const float*)d_in[3];   // [1204 x 256]
    const float* W1 = (const float*)d_in[4];   // [512 x 41]

    float* ws     = (float*)d_ws;
    float* h1     = ws;                                   // 25600*256
    float* h0     = h1 + (size_t)NB1 * HDIM;              // 1024*256
    float* agg2   = h0 + (size_t)NB0 * HDIM;              // 1024*256
    float* logits = agg2 + (size_t)NB0 * HDIM;            // 1024*41

    // big layer: 64-row tiles (512 threads, 16 waves) -> W0 L2 traffic /4
    sage_gemm_relu<10, 64><<<NB1 / 64, 512, 0, stream>>>(x1, x2, W0, h1);
    // small layer: 16-row tiles (128 threads, 4 waves)
    sage_gemm_relu<25, 16><<<NB0 / 16, 128, 0, stream>>>(x0, x1, W0, h0);
    mean25_kernel<<<(NB0 * HDIM) / 256, 256, 0, stream>>>(h1, agg2);
    dense_gemm<<<NB0 / 16, 32, 0, stream>>>(h0, agg2, W1, logits);
    logsoftmax_kernel<<<NB0 / 8, 256, 0, stream>>>(logits, (float*)d_out);
}